// RelationalGraphConvolutionNC_23983097381569
// MI455X (gfx1250) — compile-verified
//
#include <hip/hip_runtime.h>

typedef __attribute__((ext_vector_type(2))) float v2f;
typedef __attribute__((ext_vector_type(8))) float v8f;

#define N_NODES 50000
#define N_RELS  8
#define IN_DIM  128
#define OUT_DIM 128
#define N_EDGES 800000

#define TILES 5
#define ROWS  (TILES * 16)   // 80 rows per block; 50000 / 80 = 625 exactly

// out[n][d] = bias[d]
__global__ __launch_bounds__(256) void rgcn_init_out(float* __restrict__ out,
                                                     const float* __restrict__ bias) {
  int i = blockIdx.x * 256 + threadIdx.x;
  if (i < N_NODES * OUT_DIM) out[i] = bias[i & (OUT_DIM - 1)];
}

__global__ __launch_bounds__(256) void rgcn_zero_counts(int* __restrict__ counts) {
  int i = blockIdx.x * 256 + threadIdx.x;
  if (i < N_RELS * N_NODES) counts[i] = 0;
}

// counts[rel*N + tgt] += 1 per edge
__global__ __launch_bounds__(256) void rgcn_count(const int* __restrict__ triples,
                                                  int* __restrict__ counts) {
  int e = blockIdx.x * 256 + threadIdx.x;
  if (e < N_EDGES) {
    int rel = triples[3 * e + 1];
    int tgt = triples[3 * e + 2];
    atomicAdd(&counts[rel * N_NODES + tgt], 1);
  }
}

// h[r] = features @ weights[r]   (f32 WMMA 16x16x4)
// grid: (N_NODES/ROWS, N_RELS), block: 256 = 8 waves.
// Wave w owns output cols [16w, 16w+16); block owns 80 rows (5 M-tiles).
// One B fragment feeds 5 WMMAs -> 5x less weight traffic, high wmma density.
__global__ __launch_bounds__(256) void rgcn_gemm(const float* __restrict__ features,
                                                 const float* __restrict__ weights,
                                                 float* __restrict__ h) {
  __shared__ float As[ROWS][132];  // 80x128 tile, padded stride: conflict-free column reads

  const int r    = blockIdx.y;
  const int row0 = blockIdx.x * ROWS;
  const int tid  = threadIdx.x;

  // cooperative A-tile fill with float4 (b128) transfers, coalesced
  const float4* __restrict__ f4 = (const float4*)(features + (size_t)row0 * IN_DIM);
  for (int i = tid; i < ROWS * (IN_DIM / 4); i += 256) {
    int rr = i >> 5, cc = i & 31;                 // 32 float4 per row
    *(float4*)&As[rr][cc * 4] = f4[rr * 32 + cc]; // row stride 132 floats = 528B (16B aligned)
  }
  __syncthreads();

  const int wave  = tid >> 5;
  const int lane  = tid & 31;
  const int m     = lane & 15;          // row within 16-row tile / col within wave slice
  const int khalf = (lane >> 4) * 2;    // K sub-offset per ISA 16x4 fragment layout
  const int n     = wave * 16 + m;      // global output column

  const float* __restrict__ W = weights + (size_t)r * IN_DIM * OUT_DIM;

  v8f acc[TILES] = {};
  for (int kk = 0; kk < IN_DIM; kk += 4) {
    v2f b;
    b.x = W[(size_t)(kk + khalf + 0) * OUT_DIM + n];
    b.y = W[(size_t)(kk + khalf + 1) * OUT_DIM + n];
#pragma unroll
    for (int t = 0; t < TILES; ++t) {
      v2f a;
      a.x = As[t * 16 + m][kk + khalf + 0];
      a.y = As[t * 16 + m][kk + khalf + 1];
      acc[t] = __builtin_amdgcn_wmma_f32_16x16x4_f32(false, a, false, b,
                                                     (short)0, acc[t], false, false);
    }
  }

  // D layout: vgpr v -> row (v + 8*(lane>=16)), col = lane%16
  float* __restrict__ Hout = h + ((size_t)r * N_NODES + row0) * OUT_DIM;
  const int mbase = (lane >> 4) * 8;
#pragma unroll
  for (int t = 0; t < TILES; ++t) {
#pragma unroll
    for (int v = 0; v < 8; ++v) {
      Hout[(size_t)(t * 16 + mbase + v) * OUT_DIM + n] = acc[t][v];
    }
  }
}

// one wave per edge: lane handles 4 consecutive dims (float4 gather + 4 fadd atomics)
__global__ __launch_bounds__(256) void rgcn_scatter(const int* __restrict__ triples,
                                                    const int* __restrict__ counts,
                                                    const float* __restrict__ h,
                                                    float* __restrict__ out) {
  const int e    = blockIdx.x * 8 + (threadIdx.x >> 5);
  const int lane = threadIdx.x & 31;
  if (e >= N_EDGES) return;

  const int src = triples[3 * e + 0];
  const int rel = triples[3 * e + 1];
  const int tgt = triples[3 * e + 2];

  const float val = 1.0f / (float)counts[rel * N_NODES + tgt];

  const float4* __restrict__ hp =
      (const float4*)(h + ((size_t)rel * N_NODES + src) * OUT_DIM);
  float4 m4 = hp[lane];

  float* __restrict__ op = out + (size_t)tgt * OUT_DIM + lane * 4;
  unsafeAtomicAdd(op + 0, m4.x * val);
  unsafeAtomicAdd(op + 1, m4.y * val);
  unsafeAtomicAdd(op + 2, m4.z * val);
  unsafeAtomicAdd(op + 3, m4.w * val);
}

extern "C" void kernel_launch(void* const* d_in, const int* in_sizes, int n_in,
                              void* d_out, int out_size, void* d_ws, size_t ws_size,
                              hipStream_t stream) {
  const int*   triples  = (const int*)d_in[0];
  const float* features = (const float*)d_in[1];
  const float* weights  = (const float*)d_in[2];
  const float* bias     = (const float*)d_in[3];
  float*       out      = (float*)d_out;

  // ws layout: [0, 1.6MB) counts ; [2MB, 2MB+204.8MB) h
  int*   counts = (int*)d_ws;
  float* h      = (float*)((char*)d_ws + (size_t)(2u << 20));

  rgcn_init_out<<<(N_NODES * OUT_DIM + 255) / 256, 256, 0, stream>>>(out, bias);
  rgcn_zero_counts<<<(N_RELS * N_NODES + 255) / 256, 256, 0, stream>>>(counts);
  rgcn_count<<<(N_EDGES + 255) / 256, 256, 0, stream>>>(triples, counts);

  dim3 g(N_NODES / ROWS, N_RELS);
  rgcn_gemm<<<g, 256, 0, stream>>>(features, weights, h);

  rgcn_scatter<<<(N_EDGES + 7) / 8, 256, 0, stream>>>(triples, counts, h, out);
}